// VectorQuantizer_7550552507144
// MI455X (gfx1250) — compile-verified
//
#include <hip/hip_runtime.h>
#include <math.h>

// ---------------------------------------------------------------------------
// VQ-VAE vector quantizer for MI455X (gfx1250, wave32).
// distances GEMM runs on V_WMMA_F32_16X16X4_F32 (exact fp32 accumulate).
// Store-bandwidth-bound (~570MB out): distance / one-hot tiles use NT stores.
// ---------------------------------------------------------------------------

typedef __attribute__((ext_vector_type(2))) float v2f;
typedef __attribute__((ext_vector_type(8))) float v8f;

#define NROWS 131072ll      // 32*64*64 input vectors
#define DDIM  64
#define KCB   512

// flat float offsets into d_out (tuple concatenated in return order)
#define OFF_QUANT   0ll
#define OFF_PERPLEX 8388608ll        // 131072*64
#define OFF_ENC     8388609ll
#define OFF_IDX     75497473ll       // 8388609 + 131072*512
#define OFF_DIST    75628545ll       // + 131072

// ---------------------------------------------------------------------------
// Kernel 1: zero histogram (ws is NOT re-poisoned between replays) and
// precompute ||e||^2 per codebook row.
// ---------------------------------------------------------------------------
__global__ void vq_prep_kernel(const float* __restrict__ cb,
                               unsigned int* __restrict__ cnt,
                               float* __restrict__ enorm) {
    int i = threadIdx.x;                       // 512 threads, 1 block
    if (i < KCB) {
        cnt[i] = 0u;
        const float* row = cb + i * DDIM;
        float s = 0.0f;
        #pragma unroll
        for (int k = 0; k < DDIM; ++k) s = fmaf(row[k], row[k], s);
        enorm[i] = s;
    }
}

// ---------------------------------------------------------------------------
// Kernel 2: per wave = one 16-row group; 32 codebook column tiles of 16.
// ---------------------------------------------------------------------------
__global__ void __launch_bounds__(256)
vq_main_kernel(const float* __restrict__ x,
               const float* __restrict__ cb,
               const float* __restrict__ enorm,
               unsigned int* __restrict__ cnt,
               float* __restrict__ out) {
    const int lane   = threadIdx.x & 31;
    const int wave   = threadIdx.x >> 5;
    const int lane16 = lane & 15;           // = M (A frag) = N (B/D frags)
    const int hh     = lane >> 4;           // lane-half selects K pair / M+8
    const long long row0 = ((long long)blockIdx.x * 8 + wave) * 16;

    // ---- A fragments: 16x4 f32 tiles, kk = 4*f.
    // ISA layout: lane(0-15)=M, VGPR0/1 = K {0,1} (half 0) or {2,3} (half 1).
    v2f a[16];
    {
        const v2f* arow = (const v2f*)(x + (row0 + lane16) * DDIM + 2 * hh);
        #pragma unroll
        for (int f = 0; f < 16; ++f) a[f] = arow[2 * f];   // stride 4 floats
    }

    // ---- ||x||^2 per row: lane L holds half of row (L%16); pair via xor 16.
    float ps = 0.0f;
    #pragma unroll
    for (int f = 0; f < 16; ++f) {
        ps = fmaf(a[f].x, a[f].x, ps);
        ps = fmaf(a[f].y, a[f].y, ps);
    }
    const float xnfull = ps + __shfl_xor(ps, 16, 32);   // xnorm[lane%16]
    float xn[8];                                        // xnorm for D-frag row v+8*hh
    #pragma unroll
    for (int v = 0; v < 8; ++v) xn[v] = __shfl(xnfull, v + 8 * hh, 32);

    float bd[8];
    int   bi[8];
    #pragma unroll
    for (int v = 0; v < 8; ++v) { bd[v] = 3.4e38f; bi[v] = 0; }

    float* __restrict__ dist = out + OFF_DIST;

    // ---- sweep 32 column tiles of the codebook
    for (int j = 0; j < 32; ++j) {
        const int colbase = j * 16;
        const float en = enorm[colbase + lane16];          // per-lane column norm

        // B fragments 4x16 (B = codebook^T): lane%16 = N = codebook row,
        // VGPR+half = K -> two consecutive floats of that codebook row.
        const v2f* brow = (const v2f*)(cb + (colbase + lane16) * DDIM + 2 * hh);

        v8f acc = {0.f, 0.f, 0.f, 0.f, 0.f, 0.f, 0.f, 0.f};
        #pragma unroll
        for (int f = 0; f < 16; ++f) {
            v2f b = brow[2 * f];
            acc = __builtin_amdgcn_wmma_f32_16x16x4_f32(
                false, a[f], false, b, (short)0, acc, false, false);
        }

        // D layout: VGPR v -> row (v + 8*hh), col (colbase + lane%16)
        const int col = colbase + lane16;
        #pragma unroll
        for (int v = 0; v < 8; ++v) {
            float dv = fmaf(-2.0f, acc[v], xn[v] + en);
            __builtin_nontemporal_store(
                dv, dist + (row0 + v + 8 * hh) * (long long)KCB + col);
            if (dv < bd[v]) { bd[v] = dv; bi[v] = col; }   // strict < keeps lowest col
        }
    }

    // ---- argmin across the 16 lanes of each half (rows v / v+8)
    #pragma unroll
    for (int v = 0; v < 8; ++v) {
        float d = bd[v]; int i = bi[v];
        #pragma unroll
        for (int off = 8; off >= 1; off >>= 1) {
            float od = __shfl_xor(d, off, 32);
            int   oi = __shfl_xor(i, off, 32);
            if (od < d || (od == d && oi < i)) { d = od; i = oi; }
        }
        bd[v] = d; bi[v] = i;
    }

    float* __restrict__ quant = out + OFF_QUANT;
    float* __restrict__ enc   = out + OFF_ENC;     // NOTE: only 4B aligned
    float* __restrict__ eidx  = out + OFF_IDX;

    #pragma unroll
    for (int r = 0; r < 16; ++r) {
        // row r result lives in half (r>>3); read from lane 0 or 16
        const int idx_r = __shfl(bi[r & 7], (r & 8) ? 16 : 0, 32);
        const long long grow = row0 + r;

        if (lane == r) {
            eidx[grow] = (float)idx_r;
            atomicAdd(&cnt[idx_r], 1u);
        }

        // quantized = codebook[idx]; coalesced float2 per lane (8B aligned)
        {
            v2f q = ((const v2f*)(cb + (long long)idx_r * DDIM))[lane];
            ((v2f*)(quant + grow * DDIM))[lane] = q;
        }

        // one-hot row: coalesced NT b32 stores (base offset is odd -> no b128)
        const long long ebase = grow * (long long)KCB;
        #pragma unroll
        for (int t = 0; t < 16; ++t) {
            const int c = t * 32 + lane;
            __builtin_nontemporal_store((c == idx_r) ? 1.0f : 0.0f,
                                        enc + ebase + c);
        }
    }
}

// ---------------------------------------------------------------------------
// Kernel 3: perplexity = exp(-sum p*log(p+1e-10)), p = count/N
// ---------------------------------------------------------------------------
__global__ void vq_finalize_kernel(const unsigned int* __restrict__ cnt,
                                   float* __restrict__ out) {
    __shared__ float red[KCB];
    const int i = threadIdx.x;                 // 512 threads, 1 block
    float p = (float)cnt[i] * (1.0f / 131072.0f);
    red[i] = p * logf(p + 1e-10f);
    __syncthreads();
    for (int s = KCB / 2; s > 0; s >>= 1) {
        if (i < s) red[i] += red[i + s];
        __syncthreads();
    }
    if (i == 0) out[OFF_PERPLEX] = expf(-red[0]);
}

// ---------------------------------------------------------------------------
extern "C" void kernel_launch(void* const* d_in, const int* in_sizes, int n_in,
                              void* d_out, int out_size, void* d_ws, size_t ws_size,
                              hipStream_t stream) {
    const float* x  = (const float*)d_in[0];   // [32,64,64,64] f32
    const float* cb = (const float*)d_in[1];   // [512,64] f32
    float* out = (float*)d_out;

    unsigned int* cnt = (unsigned int*)d_ws;          // 512 u32
    float* enorm = (float*)d_ws + KCB;                // 512 f32

    vq_prep_kernel<<<1, 512, 0, stream>>>(cb, cnt, enorm);
    vq_main_kernel<<<1024, 256, 0, stream>>>(x, cb, enorm, cnt, out);
    vq_finalize_kernel<<<1, 512, 0, stream>>>(cnt, out);
}